// FCLayer_85804856639828
// MI455X (gfx1250) — compile-verified
//
#include <hip/hip_runtime.h>

// Problem sizes (fixed by the reference)
#define NPOS  2048
#define CIN   256
#define COUT  256
#define BATCH 64
#define KC    64            // K-chunk staged per TDM transfer

typedef __attribute__((ext_vector_type(2))) float        v2f;
typedef __attribute__((ext_vector_type(8))) float        v8f;
typedef __attribute__((ext_vector_type(4))) unsigned int v4u;
typedef __attribute__((ext_vector_type(4))) int          v4i;
typedef __attribute__((ext_vector_type(8))) int          v8i;

// LDS layout for the GEMM kernel (dynamic shared; no static LDS in that kernel,
// so these byte offsets are the raw LDS addresses the TDM writes to).
#define XS_STRIDE 260   // 256 floats + 4 pad  -> bank stride 4 (conflict-free)
#define W_STRIDE  68    // 64 floats + 4 pad   -> bank stride 4 (conflict-free)
#define LDS_X_OFF  0u
#define LDS_W0_OFF (BATCH * XS_STRIDE * 4u)                 // 66,560
#define LDS_W1_OFF (LDS_W0_OFF + COUT * W_STRIDE * 4u)      // 136,192
#define LDS_TOTAL  (LDS_W1_OFF + COUT * W_STRIDE * 4u)      // 205,824 B (< 320 KB/WGP)

#if __has_builtin(__builtin_amdgcn_tensor_load_to_lds)
#define HAS_TDM 1
#endif

#if __has_builtin(__builtin_amdgcn_s_wait_tensorcnt)
#define WAIT_TENSOR(n) __builtin_amdgcn_s_wait_tensorcnt(n)
#else
#define WAIT_TENSOR(n)
#endif

#ifdef HAS_TDM
// 2D tensor tile -> LDS via the Tensor Data Mover.
// D# per CDNA5 ISA:
//   group0: [1:0]=count=1, [63:32]=lds_addr, [120:64]=global_addr, [127:126]=type=2
//   group1: data_size=2 (4B), pad_enable/pad_interval/pad_amount, tensor_dim0/1,
//           tile_dim0/1, tensor_dim0_stride. 2D => groups 2/3 zero.
__device__ __forceinline__ void tdm_load_2d(unsigned lds_off, const void* gptr,
                                            unsigned tensor_d0, unsigned tensor_d1,
                                            unsigned tile_d0, unsigned tile_d1,
                                            unsigned d0_stride,
                                            unsigned pad_interval, unsigned pad_amount)
{
    unsigned long long ga = (unsigned long long)gptr;
    v4u g0;
    g0[0] = 1u;                                   // count=1 (valid), no gather
    g0[1] = lds_off;                              // lds_addr (bytes)
    g0[2] = (unsigned)ga;                         // global_addr[31:0]
    g0[3] = (unsigned)((ga >> 32) & 0x01FFFFFFu)  // global_addr[56:32]
          | 0x80000000u;                          // type=2 ("image")
    v8i g1;
    g1[0] = (int)((2u << 16)                      // data_size = 4 bytes
                | (1u << 20)                      // pad_enable
                | ((pad_interval & 7u) << 22)
                | ((pad_amount & 127u) << 25));
    g1[1] = (int)((tensor_d0 & 0xFFFFu) << 16);                                  // dim0 lo16
    g1[2] = (int)(((tensor_d0 >> 16) & 0xFFFFu) | ((tensor_d1 & 0xFFFFu) << 16));
    g1[3] = (int)(((tensor_d1 >> 16) & 0xFFFFu) | ((tile_d0 & 0xFFFFu) << 16));
    g1[4] = (int)(tile_d1 & 0xFFFFu);             // tile_dim1; tile_dim2=0 (2D)
    g1[5] = (int)d0_stride;                       // tensor_dim0_stride[31:0]
    g1[6] = 0;
    g1[7] = 0;
    v4i gz = {0, 0, 0, 0};
#if defined(__clang_major__) && (__clang_major__ >= 23)
    v8i gz8 = {0, 0, 0, 0, 0, 0, 0, 0};
    __builtin_amdgcn_tensor_load_to_lds(g0, g1, gz, gz, gz8, 0);
#else
    __builtin_amdgcn_tensor_load_to_lds(g0, g1, gz, gz, 0);
#endif
}
#endif // HAS_TDM

// ---- fallback staging (only used if the TDM builtin is unavailable) ----
__device__ __forceinline__ void stage_x_direct(float* xs, const float* xg, int tid)
{
    for (int idx = tid; idx < BATCH * CIN; idx += 256) {
        int r = idx >> 8, c = idx & 255;
        xs[r * XS_STRIDE + c] = xg[idx];
    }
}
__device__ __forceinline__ void stage_w_direct(float* wbuf, const float* wg, int tid)
{
    for (int idx = tid; idx < COUT * KC; idx += 256) {
        int r = idx >> 6, c = idx & 63;
        wbuf[r * W_STRIDE + c] = wg[r * CIN + c];
    }
}

// One K-chunk of the per-position GEMM.
// A (16x4 f32): lanes 0-15 hold M rows, VGPR0/1 = K pair; lanes 16-31 = K+2/K+3.
// B (4x16 f32): mirrored K split; lane holds W[o = n_base + lane%16][k].
__device__ __forceinline__ void compute_chunk(const float* xs, const float* wb,
                                              int ccol, int n0, int l16, int lhalf,
                                              v8f (&acc)[4][2])
{
#pragma unroll
    for (int kk = 0; kk < KC; kk += 4) {
        v2f a[4];
#pragma unroll
        for (int mt = 0; mt < 4; ++mt)
            a[mt] = *(const v2f*)&xs[(mt * 16 + l16) * XS_STRIDE + ccol + kk + 2 * lhalf];
        v2f bq[2];
#pragma unroll
        for (int nt = 0; nt < 2; ++nt)
            bq[nt] = *(const v2f*)&wb[(n0 + nt * 16 + l16) * W_STRIDE + kk + 2 * lhalf];
#pragma unroll
        for (int mt = 0; mt < 4; ++mt)
#pragma unroll
            for (int nt = 0; nt < 2; ++nt)
                acc[mt][nt] = __builtin_amdgcn_wmma_f32_16x16x4_f32(
                    false, a[mt], false, bq[nt], (short)0, acc[mt][nt], false, false);
    }
}

// One workgroup per position i. 8 waves; wave w owns N columns [32w, 32w+32),
// all 4 M(=batch) tiles -> 8 accumulator tiles of 16x16 f32 per wave.
__global__ __launch_bounds__(256) void pos_gemm_wmma(
    const float* __restrict__ xA,   // [NPOS][BATCH][CIN]
    const float* __restrict__ Wm,   // [NPOS][COUT][CIN]
    const float* __restrict__ bias, // [NPOS][COUT]
    float* __restrict__ outT)       // [NPOS][BATCH][COUT]
{
    extern __shared__ float lds[];
    float* xs = lds;                        // [BATCH][XS_STRIDE]
    float* w0 = lds + BATCH * XS_STRIDE;    // [COUT][W_STRIDE]
    float* w1 = w0 + COUT * W_STRIDE;       // [COUT][W_STRIDE]

    const int i     = blockIdx.x;
    const int tid   = threadIdx.x;
    const int wave  = tid >> 5;
    const int lane  = tid & 31;
    const int l16   = lane & 15;
    const int lhalf = lane >> 4;
    const int n0    = wave * 32;

    const float* xg = xA + (size_t)i * (BATCH * CIN);
    const float* wg = Wm + (size_t)i * (COUT * CIN);

    v8f acc[4][2] = {};

#ifdef HAS_TDM
    if (wave == 0) {
        // x slab: rows=BATCH, row=256 DW, pad 4 DW -> stride 260 (interval 7=256DW, amount 3=4DW)
        tdm_load_2d(LDS_X_OFF, xg, CIN, BATCH, CIN, BATCH, CIN, 7, 3);
        // W chunks: rows=COUT, row=64 DW, pad 4 DW -> stride 68 (interval 5=64DW, amount 3=4DW)
        tdm_load_2d(LDS_W0_OFF, wg + 0 * KC, CIN, COUT, KC, COUT, CIN, 5, 3);
        tdm_load_2d(LDS_W1_OFF, wg + 1 * KC, CIN, COUT, KC, COUT, CIN, 5, 3);
        WAIT_TENSOR(1);                 // in-order: x and W-chunk0 complete
    }
    __syncthreads();
    compute_chunk(xs, w0, 0 * KC, n0, l16, lhalf, acc);
    __syncthreads();                    // everyone done reading w0
    if (wave == 0) { tdm_load_2d(LDS_W0_OFF, wg + 2 * KC, CIN, COUT, KC, COUT, CIN, 5, 3); WAIT_TENSOR(1); }
    __syncthreads();                    // W-chunk1 ready
    compute_chunk(xs, w1, 1 * KC, n0, l16, lhalf, acc);
    __syncthreads();
    if (wave == 0) { tdm_load_2d(LDS_W1_OFF, wg + 3 * KC, CIN, COUT, KC, COUT, CIN, 5, 3); WAIT_TENSOR(1); }
    __syncthreads();                    // W-chunk2 ready
    compute_chunk(xs, w0, 2 * KC, n0, l16, lhalf, acc);
    __syncthreads();
    if (wave == 0) { WAIT_TENSOR(0); }  // W-chunk3 ready
    __syncthreads();
    compute_chunk(xs, w1, 3 * KC, n0, l16, lhalf, acc);
#else
    stage_x_direct(xs, xg, tid);
    stage_w_direct(w0, wg + 0 * KC, tid);
    stage_w_direct(w1, wg + 1 * KC, tid);
    __syncthreads();
    compute_chunk(xs, w0, 0 * KC, n0, l16, lhalf, acc);
    __syncthreads();
    stage_w_direct(w0, wg + 2 * KC, tid);
    __syncthreads();
    compute_chunk(xs, w1, 1 * KC, n0, l16, lhalf, acc);
    __syncthreads();
    stage_w_direct(w1, wg + 3 * KC, tid);
    __syncthreads();
    compute_chunk(xs, w0, 2 * KC, n0, l16, lhalf, acc);
    compute_chunk(xs, w1, 3 * KC, n0, l16, lhalf, acc);
#endif

    // Epilogue: fused bias add, contiguous stores to outT[i][b][o].
    // C/D layout: VGPR r holds (M = mt*16 + r + 8*(lane>>4), N = n_base + lane&15).
#pragma unroll
    for (int nt = 0; nt < 2; ++nt) {
        const int o  = n0 + nt * 16 + l16;
        const float bb = bias[(size_t)i * COUT + o];
#pragma unroll
        for (int mt = 0; mt < 4; ++mt)
#pragma unroll
            for (int r = 0; r < 8; ++r) {
                const int m = mt * 16 + r + 8 * lhalf;
                outT[(size_t)i * (BATCH * COUT) + (size_t)m * COUT + o] = acc[mt][nt][r] + bb;
            }
    }
}

// Tiled f32 transpose: dst[c][r] = src[r][c]; src is R x C. Grid (C/32, R/32), block (32,8).
__global__ __launch_bounds__(256) void transpose_f32(
    float* __restrict__ dst, const float* __restrict__ src, int R, int C)
{
    __shared__ float tile[32][33];
    const int cb = blockIdx.x * 32, rb = blockIdx.y * 32;
    const int tx = threadIdx.x, ty = threadIdx.y;
#pragma unroll
    for (int j = 0; j < 32; j += 8)
        tile[ty + j][tx] = src[(size_t)(rb + ty + j) * C + (cb + tx)];
    __syncthreads();
#pragma unroll
    for (int j = 0; j < 32; j += 8)
        dst[(size_t)(cb + ty + j) * R + (rb + tx)] = tile[tx][ty + j];
}

extern "C" void kernel_launch(void* const* d_in, const int* in_sizes, int n_in,
                              void* d_out, int out_size, void* d_ws, size_t ws_size,
                              hipStream_t stream)
{
    const float* x  = (const float*)d_in[0];   // [BATCH][CIN][NPOS]
    const float* Wm = (const float*)d_in[1];   // [NPOS][COUT][CIN]
    const float* b  = (const float*)d_in[2];   // [NPOS][COUT]
    float* out = (float*)d_out;                // [BATCH][COUT][NPOS]

    float* xA   = (float*)d_ws;                              // [NPOS][BATCH][CIN]  (128 MB)
    float* outT = xA + (size_t)NPOS * BATCH * CIN;           // [NPOS][BATCH][COUT] (128 MB)

    dim3 tb(32, 8);
    // 1) xA[i][b][c] = x[b][c][i]  (src viewed as [BATCH*CIN][NPOS])
    transpose_f32<<<dim3(NPOS / 32, (BATCH * CIN) / 32), tb, 0, stream>>>(
        xA, x, BATCH * CIN, NPOS);

    // 2) per-position WMMA GEMM with TDM-staged operands
    pos_gemm_wmma<<<dim3(NPOS), dim3(256), LDS_TOTAL, stream>>>(xA, Wm, b, outT);

    // 3) out[b][o][i] = outT[i][b][o]  (src viewed as [NPOS][BATCH*COUT])
    transpose_f32<<<dim3((BATCH * COUT) / 32, NPOS / 32), tb, 0, stream>>>(
        out, outT, NPOS, BATCH * COUT);
}